// CausalSelfAttention_17686675325189
// MI455X (gfx1250) — compile-verified
//
#include <hip/hip_runtime.h>
#include <math.h>

typedef __attribute__((ext_vector_type(2))) float v2f;
typedef __attribute__((ext_vector_type(4))) float fv4;
typedef __attribute__((ext_vector_type(8))) float v8f;
typedef __attribute__((ext_vector_type(4))) int v4i;

#define BB  2
#define TT  2048
#define CC  1024
#define HH  16
#define DHH 64

#define AS1 __attribute__((address_space(1)))
#define AS3 __attribute__((address_space(3)))

static __device__ __forceinline__ v8f wmma_f32(v2f a, v2f b, v8f c) {
  return __builtin_amdgcn_wmma_f32_16x16x4_f32(false, a, false, b, (short)0, c,
                                               false, false);
}

// Async global -> LDS copy of 16 bytes per lane (tracked by ASYNCcnt).
static __device__ __forceinline__ void async_ld_b128(const void* g, void* l) {
#if __has_builtin(__builtin_amdgcn_global_load_async_to_lds_b128)
  __builtin_amdgcn_global_load_async_to_lds_b128(
      (AS1 v4i*)(AS1 void*)(void*)g, (AS3 v4i*)(AS3 void*)l, 0, 0);
#else
  unsigned loff = (unsigned)(unsigned long)(AS3 void*)l;
  asm volatile("global_load_async_to_lds_b128 %0, %1, off" ::"v"(loff), "v"(g)
               : "memory");
#endif
}

template <int N>
static __device__ __forceinline__ void wait_async() {
#if __has_builtin(__builtin_amdgcn_s_wait_asynccnt)
  __builtin_amdgcn_s_wait_asynccnt(N);
#else
  asm volatile("s_wait_asynccnt %0" ::"i"(N) : "memory");
#endif
}

// ---------------------------------------------------------------------------
// GEMM: C[M,N] = A[M,K] * B  (+bias).  BT=false: B is [K,N] row-major.
//                                      BT=true : B is [N,K] row-major (B^T).
// Block tile 128x64, K-chunk 32, 8 waves, wave tile 32x32.
// Async-staged LDS with double buffering: prefetch chunk i+1 while
// computing chunk i; s_wait_asynccnt 6 retires chunk i's 6 copies/thread.
// ---------------------------------------------------------------------------
template <bool BT, bool BIAS>
__global__ __launch_bounds__(256) void gemm_f32_wmma(
    const float* __restrict__ A, const float* __restrict__ Bm,
    const float* __restrict__ bias, float* __restrict__ Cm,
    int M, int N, int K) {
  constexpr int TM = 128, TN = 64, KC = 32;
  constexpr int LDA = KC + 4;                    // 36 floats (144B rows)
  constexpr int LDB = BT ? (KC + 4) : (TN + 4);  // [n][k] 36 : [k][n] 68
  constexpr int ASIZE = TM * LDA;                // 4608 floats
  constexpr int BSIZE = BT ? TN * LDB : KC * LDB;
  __shared__ float lA[2 * ASIZE];
  __shared__ float lB[2 * BSIZE];

  const int tid  = threadIdx.x;
  const int lane = tid & 31;
  const int w    = tid >> 5;   // 0..7
  const int wm   = w >> 1;     // 0..3
  const int wn   = w & 1;      // 0..1
  const int half = lane >> 4;  // 0/1
  const int l15  = lane & 15;
  const int m0   = blockIdx.y * TM;
  const int n0   = blockIdx.x * TN;

  v8f acc[2][2] = {};

  auto stageA = [&](int k0, int buf) {
#pragma unroll
    for (int i = 0; i < 4; ++i) {
      int idx = tid + i * 256;
      int row = idx >> 3;  // 8 float4 per row
      int col = (idx & 7) << 2;
      async_ld_b128(A + (size_t)(m0 + row) * K + k0 + col,
                    &lA[buf * ASIZE + row * LDA + col]);
    }
  };
  auto stageB = [&](int k0, int buf) {
    if constexpr (!BT) {
      // B[K,N]: 32 rows x 64 cols = 512 float4, 2 per thread
#pragma unroll
      for (int i = 0; i < 2; ++i) {
        int idx = tid + i * 256;
        int row = idx >> 4;  // 16 float4 per row
        int col = (idx & 15) << 2;
        async_ld_b128(Bm + (size_t)(k0 + row) * N + n0 + col,
                      &lB[buf * BSIZE + row * LDB + col]);
      }
    } else {
      // Bm[N,K]: stage untransposed as [n][k]; frag read handles the rest
#pragma unroll
      for (int i = 0; i < 2; ++i) {
        int idx  = tid + i * 256;
        int nrow = idx >> 3;  // 8 float4 per n-row
        int kcol = (idx & 7) << 2;
        async_ld_b128(Bm + (size_t)(n0 + nrow) * K + k0 + kcol,
                      &lB[buf * BSIZE + nrow * LDB + kcol]);
      }
    }
  };

  const int NKC = K / KC;
  stageA(0, 0);
  stageB(0, 0);

  for (int ic = 0; ic < NKC; ++ic) {
    const int buf = ic & 1;
    if (ic + 1 < NKC) {
      stageA((ic + 1) * KC, buf ^ 1);  // prefetch into other buffer
      stageB((ic + 1) * KC, buf ^ 1);
      wait_async<6>();  // chunk ic's 6 copies retired (in-order)
    } else {
      wait_async<0>();
    }
    __syncthreads();

    const float* bufA = &lA[buf * ASIZE];
    const float* bufB = &lB[buf * BSIZE];
#pragma unroll
    for (int kk = 0; kk < KC; kk += 4) {
      const int koff = kk + (half << 1);  // lanes<16: K=0,1 ; lanes>=16: K=2,3
      v2f af[2], bf[2];
#pragma unroll
      for (int i = 0; i < 2; ++i) {
        const float* p = &bufA[(wm * 32 + i * 16 + l15) * LDA + koff];
        af[i].x = p[0];
        af[i].y = p[1];
      }
#pragma unroll
      for (int j = 0; j < 2; ++j) {
        if constexpr (!BT) {
          bf[j].x = bufB[(koff + 0) * LDB + wn * 32 + j * 16 + l15];
          bf[j].y = bufB[(koff + 1) * LDB + wn * 32 + j * 16 + l15];
        } else {
          const float* p = &bufB[(wn * 32 + j * 16 + l15) * LDB + koff];
          bf[j].x = p[0];
          bf[j].y = p[1];
        }
      }
#pragma unroll
      for (int i = 0; i < 2; ++i)
#pragma unroll
        for (int j = 0; j < 2; ++j)
          acc[i][j] = wmma_f32(af[i], bf[j], acc[i][j]);
    }
    __syncthreads();  // protects the buffer the next prefetch writes
  }

  // epilogue: C/D layout -> VGPR e holds row e (lanes 0-15) / e+8 (lanes 16-31)
#pragma unroll
  for (int i = 0; i < 2; ++i) {
#pragma unroll
    for (int j = 0; j < 2; ++j) {
      int ncol = n0 + wn * 32 + j * 16 + l15;
      float bv = BIAS ? bias[ncol] : 0.0f;
#pragma unroll
      for (int e = 0; e < 8; ++e) {
        int mrow = m0 + wm * 32 + i * 16 + e + (half << 3);
        Cm[(size_t)mrow * N + ncol] = acc[i][j][e] + bv;
      }
    }
  }
}

// ---------------------------------------------------------------------------
// Flash-style causal attention.  Q/K/V in (B,T,H*DH) layout.
// Grid: (T/64, B*H), block = 128 threads (4 waves x 16 query rows).
// K/V chunks staged with async LDS copies.
// ---------------------------------------------------------------------------
__global__ __launch_bounds__(128) void attn_kernel(
    const float* __restrict__ Q, const float* __restrict__ Kt,
    const float* __restrict__ Vt, float* __restrict__ Y) {
  constexpr int LDK = DHH + 4;  // 68 floats
  constexpr int LDP = 20;       // P scratch stride
  __shared__ float lK[64 * LDK];
  __shared__ float lV[64 * LDK];
  __shared__ float lP[4 * 16 * LDP];

  const int tid  = threadIdx.x;
  const int lane = tid & 31;
  const int w    = tid >> 5;  // 0..3
  const int half = lane >> 4;
  const int l15  = lane & 15;
  const int qb   = blockIdx.x * 64;
  const int bh   = blockIdx.y;
  const int b    = bh / HH;
  const int h    = bh % HH;
  const size_t baseBH = ((size_t)b * TT) * CC + (size_t)h * DHH;

  const int qw   = qb + w * 16;  // wave's first query row
  const int qwhi = __builtin_amdgcn_readfirstlane(qw + 15);
  const float scale = 0.125f;  // 1/sqrt(64)

  // Q rows as A-fragments (pre-scaled): 16 k-steps of 4 over DH=64
  v2f qf[16];
  {
    const float* qp = Q + baseBH + (size_t)(qw + l15) * CC;
#pragma unroll
    for (int ds = 0; ds < 16; ++ds) {
      int d = ds * 4 + (half << 1);
      qf[ds].x = qp[d] * scale;
      qf[ds].y = qp[d + 1] * scale;
    }
  }

  v8f o[4] = {};
  float rowM[8], rowL[8];
#pragma unroll
  for (int e = 0; e < 8; ++e) { rowM[e] = -INFINITY; rowL[e] = 0.0f; }

  float* lPw = &lP[w * 16 * LDP];

  for (int sb = 0; sb <= qb; sb += 64) {
    // stage 64 keys + 64 values: 16 async b128 copies per thread
#pragma unroll
    for (int i = 0; i < 8; ++i) {
      int idx = tid + i * 128;
      int row = idx >> 4;
      int col = (idx & 15) << 2;
      size_t g = baseBH + (size_t)(sb + row) * CC + col;
      async_ld_b128(Kt + g, &lK[row * LDK + col]);
      async_ld_b128(Vt + g, &lV[row * LDK + col]);
    }
    wait_async<0>();
    __syncthreads();

#pragma unroll 1
    for (int j = 0; j < 4; ++j) {
      const int st = sb + j * 16;  // key sub-tile base (wave-uniform)
      if (st > qwhi) break;        // fully masked for this wave

      // S = (Q*scale) K^T  : B[k=d][n=s] = K[s][d]
      v8f s = {};
#pragma unroll
      for (int ds = 0; ds < 16; ++ds) {
        int dk = ds * 4 + (half << 1);
        v2f bf;
        bf.x = lK[(j * 16 + l15) * LDK + dk];
        bf.y = lK[(j * 16 + l15) * LDK + dk + 1];
        s = wmma_f32(qf[ds], bf, s);
      }

      // causal mask (reference: mask AFTER scaling, to -1e10)
      float p[8];
#pragma unroll
      for (int e = 0; e < 8; ++e) {
        int qrow = qw + e + (half << 3);
        int scol = st + l15;
        float val = s[e];
        if (scol > qrow) val = -1e10f;
        p[e] = val;
      }

      // online softmax: 16-lane row reductions (xor 8,4,2,1 stays in half)
#pragma unroll
      for (int e = 0; e < 8; ++e) {
        float mx = p[e];
#pragma unroll
        for (int off = 8; off >= 1; off >>= 1)
          mx = fmaxf(mx, __shfl_xor(mx, off, 32));
        float newM = fmaxf(rowM[e], mx);
        float c    = __expf(rowM[e] - newM);
        rowM[e]    = newM;
        float pe   = __expf(p[e] - newM);
        p[e]       = pe;
        float sum  = pe;
#pragma unroll
        for (int off = 8; off >= 1; off >>= 1)
          sum += __shfl_xor(sum, off, 32);
        rowL[e] = rowL[e] * c + sum;
#pragma unroll
        for (int cx = 0; cx < 4; ++cx) o[cx][e] *= c;
      }

      // C-layout -> A-layout bounce through per-wave LDS scratch
#pragma unroll
      for (int e = 0; e < 8; ++e)
        lPw[(e + (half << 3)) * LDP + l15] = p[e];

      // O += P * V  : A = P (16x16), B[k=s][n=d] = V[s][d]
#pragma unroll
      for (int ks = 0; ks < 4; ++ks) {
        int koff = ks * 4 + (half << 1);
        v2f af;
        af.x = lPw[l15 * LDP + koff];
        af.y = lPw[l15 * LDP + koff + 1];
#pragma unroll
        for (int cx = 0; cx < 4; ++cx) {
          v2f bf;
          bf.x = lV[(j * 16 + koff) * LDK + cx * 16 + l15];
          bf.y = lV[(j * 16 + koff + 1) * LDK + cx * 16 + l15];
          o[cx] = wmma_f32(af, bf, o[cx]);
        }
      }
    }
    __syncthreads();
  }

  // normalize + store into (B,T,C) layout (== transpose(0,2,1,3).reshape)
#pragma unroll
  for (int cx = 0; cx < 4; ++cx) {
#pragma unroll
    for (int e = 0; e < 8; ++e) {
      int m = qw + e + (half << 3);
      Y[((size_t)b * TT + m) * CC + h * DHH + cx * 16 + l15] =
          o[cx][e] / rowL[e];
    }
  }
}

// ---------------------------------------------------------------------------
extern "C" void kernel_launch(void* const* d_in, const int* in_sizes, int n_in,
                              void* d_out, int out_size, void* d_ws,
                              size_t ws_size, hipStream_t stream) {
  (void)in_sizes; (void)n_in; (void)out_size; (void)ws_size;
  const float* x  = (const float*)d_in[0];
  const float* Wq = (const float*)d_in[1];
  const float* Wk = (const float*)d_in[2];
  const float* Wv = (const float*)d_in[3];
  const float* Wp = (const float*)d_in[4];
  const float* bp = (const float*)d_in[5];
  float* out = (float*)d_out;

  const size_t MAT = (size_t)BB * TT * CC;  // 4Mi floats = 16 MB
  float* q = (float*)d_ws;
  float* k = q + MAT;
  float* v = k + MAT;
  float* y = v + MAT;

  dim3 gg(CC / 64, (BB * TT) / 128);  // (16, 32)
  gemm_f32_wmma<false, false><<<gg, 256, 0, stream>>>(x, Wq, nullptr, q,
                                                      BB * TT, CC, CC);
  gemm_f32_wmma<false, false><<<gg, 256, 0, stream>>>(x, Wk, nullptr, k,
                                                      BB * TT, CC, CC);
  gemm_f32_wmma<false, false><<<gg, 256, 0, stream>>>(x, Wv, nullptr, v,
                                                      BB * TT, CC, CC);
  attn_kernel<<<dim3(TT / 64, BB * HH), 128, 0, stream>>>(q, k, v, y);
  gemm_f32_wmma<true, true><<<gg, 256, 0, stream>>>(y, Wp, bp, out,
                                                    BB * TT, CC, CC);
}